// GcnUnit_40157944217632
// MI455X (gfx1250) — compile-verified
//
#include <hip/hip_runtime.h>
#include <hip/hip_bf16.h>

#define NN 100000
#define EE 1600000
#define CC 128
#define GN_EPS 1e-5f

typedef float v2f __attribute__((ext_vector_type(2)));
typedef float v8f __attribute__((ext_vector_type(8)));

// ---------------------------------------------------------------------------
// GEMM: C[M,128] = A[M,128] @ B[128,128], fp32 WMMA 16x16x4, one wave per tile
// ---------------------------------------------------------------------------
__global__ void gemm_wmma_f32(const float* __restrict__ A,
                              const float* __restrict__ B,
                              float* __restrict__ C, int M) {
  int wave = (int)((blockIdx.x * blockDim.x + threadIdx.x) >> 5);
  int lane = threadIdx.x & 31;
  int tile_m = wave >> 3;        // 8 N-tiles (128/16)
  int tile_n = wave & 7;
  if (tile_m * 16 >= M) return;
  int m0 = tile_m * 16, n0 = tile_n * 16;
  int half = lane >> 4;          // 0: K pair {k,k+1}; 1: {k+2,k+3}
  int l15  = lane & 15;

  v8f acc = {};
  const float* arow = A + (size_t)(m0 + l15) * CC;
  for (int k = 0; k < CC; k += 4) {
    int ka = k + half * 2;
    v2f a = *(const v2f*)(arow + ka);                 // A[m0+l15, ka..ka+1]
    v2f b;
    b.x = B[(size_t)ka * CC + n0 + l15];              // B[ka,   n0+l15]
    b.y = B[(size_t)(ka + 1) * CC + n0 + l15];        // B[ka+1, n0+l15]
    acc = __builtin_amdgcn_wmma_f32_16x16x4_f32(false, a, false, b,
                                                (short)0, acc, false, false);
  }
#pragma unroll
  for (int v = 0; v < 8; ++v)
    C[(size_t)(m0 + v + half * 8) * CC + n0 + l15] = acc[v];
}

// ---------------------------------------------------------------------------
// Degree / init helpers
// ---------------------------------------------------------------------------
__global__ void init_deg(float* deg) {
  int n = blockIdx.x * blockDim.x + threadIdx.x;
  if (n < NN) deg[n] = 1.0f;                         // self-loop
}
__global__ void count_deg(const long long* __restrict__ dst, float* deg) {
  int e = blockIdx.x * blockDim.x + threadIdx.x;
  if (e < EE) atomicAdd(&deg[(int)dst[e]], 1.0f);
}
__global__ void node_dinv(float* deg) {
  int n = blockIdx.x * blockDim.x + threadIdx.x;
  if (n < NN) deg[n] = rsqrtf(deg[n]);
}
// t[n,c] = h[n,c]*dinv[n]^2 + b1[c]   (self-loop term + bias)
__global__ void gcn_init_t(const float* __restrict__ h, const float* __restrict__ dinv,
                           const float* __restrict__ b1, float* __restrict__ t) {
  size_t idx = (size_t)blockIdx.x * blockDim.x + threadIdx.x;
  if (idx >= (size_t)NN * CC) return;
  int n = (int)(idx >> 7), c = (int)(idx & 127);
  float di = dinv[n];
  t[idx] = h[idx] * di * di + b1[c];
}

// one wave per edge: t[dst] += h[src] * dinv[src]*dinv[dst]
__global__ void gcn_scatter(const long long* __restrict__ src, const long long* __restrict__ dst,
                            const float* __restrict__ h, const float* __restrict__ dinv,
                            float* __restrict__ t) {
  int w = (int)((blockIdx.x * blockDim.x + threadIdx.x) >> 5);
  if (w >= EE) return;
  int lane = threadIdx.x & 31;
  int s = (int)src[w], d = (int)dst[w];
  float nrm = dinv[s] * dinv[d];
  float4 hv = *(const float4*)(h + (size_t)s * CC + lane * 4);
  float* tp = t + (size_t)d * CC + lane * 4;
  atomicAdd(tp + 0, hv.x * nrm);
  atomicAdd(tp + 1, hv.y * nrm);
  atomicAdd(tp + 2, hv.z * nrm);
  atomicAdd(tp + 3, hv.w * nrm);
}

// ---------------------------------------------------------------------------
// GraphNorm moments (single pass: sum & sumsq per channel)
// ---------------------------------------------------------------------------
__global__ void zero_moms(float* sums, float* sumsq) {
  int c = threadIdx.x;
  if (c < CC) { sums[c] = 0.f; sumsq[c] = 0.f; }
}
__global__ void moments(const float* __restrict__ t, const float* __restrict__ denom,
                        const float* __restrict__ bias, float* sums, float* sumsq,
                        int rows_per_block) {
  int c = threadIdx.x;                               // 128 threads
  int r0 = blockIdx.x * rows_per_block;
  int r1 = min(r0 + rows_per_block, NN);
  float bc = bias ? bias[c] : 0.f;
  float s = 0.f, s2 = 0.f;
  for (int n = r0; n < r1; ++n) {
    float v = t[(size_t)n * CC + c];
    if (denom) v = v / denom[n] + bc;
    s += v; s2 += v * v;
  }
  atomicAdd(&sums[c], s);
  atomicAdd(&sumsq[c], s2);
}
// out[n,c] = xin[n,c] + leaky01( gw*(v - ms*mean)*rsqrt(var+eps) + gb )
__global__ void norm_out(const float* __restrict__ t, const float* __restrict__ denom,
                         const float* __restrict__ bias, const float* __restrict__ xin,
                         const float* __restrict__ sums, const float* __restrict__ sumsq,
                         const float* __restrict__ gw, const float* __restrict__ gb,
                         const float* __restrict__ gms, float* __restrict__ out) {
  size_t idx = (size_t)blockIdx.x * blockDim.x + threadIdx.x;
  if (idx >= (size_t)NN * CC) return;
  int n = (int)(idx >> 7), c = (int)(idx & 127);
  float v = t[idx];
  if (denom) v = v / denom[n] + bias[c];
  const float invN = 1.0f / (float)NN;
  float mean = sums[c] * invN;
  float q    = sumsq[c] * invN;
  float ms   = gms[c];
  float z    = v - ms * mean;
  float var  = q - 2.f * ms * mean * mean + ms * ms * mean * mean;
  var = fmaxf(var, 0.f);
  float y = gw[c] * z * rsqrtf(var + GN_EPS) + gb[c];
  y = y >= 0.f ? y : 0.01f * y;
  out[idx] = xin[idx] + y;
}

// ---------------------------------------------------------------------------
// GAT
// ---------------------------------------------------------------------------
__device__ __forceinline__ void atomicMaxF(float* addr, float v) {
  if (v >= 0.f) atomicMax((int*)addr, __float_as_int(v));
  else          atomicMin((unsigned int*)addr, __float_as_uint(v));
}

// per node: a_s, a_d, self logit, m init
__global__ void attn_node(const float* __restrict__ h, const float* __restrict__ att_s,
                          const float* __restrict__ att_d, float* a_s, float* a_d,
                          float* eself, float* m) {
  int n = blockIdx.x, c = threadIdx.x;               // 128 threads
  float hv = h[(size_t)n * CC + c];
  __shared__ float ss[CC], sd[CC];
  ss[c] = hv * att_s[c];
  sd[c] = hv * att_d[c];
  __syncthreads();
  for (int off = 64; off > 0; off >>= 1) {
    if (c < off) { ss[c] += ss[c + off]; sd[c] += sd[c + off]; }
    __syncthreads();
  }
  if (c == 0) {
    float as_ = ss[0], ad_ = sd[0];
    a_s[n] = as_; a_d[n] = ad_;
    float e = as_ + ad_;
    e = e >= 0.f ? e : 0.2f * e;                     // leaky 0.2
    eself[n] = e; m[n] = e;                          // self-loop seeds segment max
  }
}
__global__ void gat_max(const long long* __restrict__ src, const long long* __restrict__ dst,
                        const float* __restrict__ a_s, const float* __restrict__ a_d,
                        float* __restrict__ ebuf, float* __restrict__ m) {
  int e = blockIdx.x * blockDim.x + threadIdx.x;
  if (e >= EE) return;
  int s = (int)src[e], d = (int)dst[e];
  float x = a_s[s] + a_d[d];
  x = x >= 0.f ? x : 0.2f * x;
  ebuf[e] = x;
  atomicMaxF(&m[d], x);
}
// init denom + t with self-loop contribution
__global__ void gat_init(const float* __restrict__ h, const float* __restrict__ eself,
                         const float* __restrict__ m, float* __restrict__ t,
                         float* __restrict__ denom) {
  size_t idx = (size_t)blockIdx.x * blockDim.x + threadIdx.x;
  if (idx >= (size_t)NN * CC) return;
  int n = (int)(idx >> 7), c = (int)(idx & 127);
  float p = __expf(eself[n] - m[n]);
  t[idx] = h[idx] * p;
  if (c == 0) denom[n] = p;
}
// one wave per edge: denom[dst]+=ex ; t[dst] += h[src]*ex (normalize later)
__global__ void gat_scatter(const long long* __restrict__ src, const long long* __restrict__ dst,
                            const float* __restrict__ h, const float* __restrict__ ebuf,
                            const float* __restrict__ m, float* __restrict__ t,
                            float* __restrict__ denom) {
  int w = (int)((blockIdx.x * blockDim.x + threadIdx.x) >> 5);
  if (w >= EE) return;
  int lane = threadIdx.x & 31;
  int s = (int)src[w], d = (int)dst[w];
  float ex = __expf(ebuf[w] - m[d]);
  if (lane == 0) atomicAdd(&denom[d], ex);
  float4 hv = *(const float4*)(h + (size_t)s * CC + lane * 4);
  float* tp = t + (size_t)d * CC + lane * 4;
  atomicAdd(tp + 0, hv.x * ex);
  atomicAdd(tp + 1, hv.y * ex);
  atomicAdd(tp + 2, hv.z * ex);
  atomicAdd(tp + 3, hv.w * ex);
}

// ---------------------------------------------------------------------------
extern "C" void kernel_launch(void* const* d_in, const int* in_sizes, int n_in,
                              void* d_out, int out_size, void* d_ws, size_t ws_size,
                              hipStream_t stream) {
  const float*     x    = (const float*)d_in[0];
  const long long* edges= (const long long*)d_in[1];
  const float*     W1   = (const float*)d_in[2];
  const float*     b1   = (const float*)d_in[3];
  const float*     gw   = (const float*)d_in[4];
  const float*     gb   = (const float*)d_in[5];
  const float*     gms  = (const float*)d_in[6];
  const float*     Wg   = (const float*)d_in[7];
  const float*     bg   = (const float*)d_in[8];
  const float*     atts = (const float*)d_in[9];
  const float*     attd = (const float*)d_in[10];
  float* out = (float*)d_out;

  const long long* srcp = edges;
  const long long* dstp = edges + EE;

  // workspace layout (floats)
  float* ws    = (float*)d_ws;
  float* h     = ws;                       // N*C
  float* t     = h    + (size_t)NN * CC;   // N*C
  float* xmid  = t    + (size_t)NN * CC;   // N*C
  float* deg   = xmid + (size_t)NN * CC;   // N (becomes dinv)
  float* a_s   = deg  + NN;                // N
  float* a_d   = a_s  + NN;                // N
  float* eself = a_d  + NN;                // N
  float* mbuf  = eself+ NN;                // N
  float* den   = mbuf + NN;                // N
  float* ebuf  = den  + NN;                // E
  float* sums  = ebuf + EE;                // C
  float* sumsq = sums + CC;                // C

  const int BT = 256;
  dim3 b(BT);
  int gN   = (NN + BT - 1) / BT;                 // node-wise
  int gE   = (EE + BT - 1) / BT;                 // edge-wise (thread per edge)
  int gEW  = (int)(((size_t)EE * 32 + BT - 1) / BT); // edge-wise (wave per edge)
  int gNC  = (int)(((size_t)NN * CC + BT - 1) / BT); // element-wise
  int gGEMM = ((NN / 16) * 8 * 32) / BT;         // wave per 16x16 tile
  const int RPB = 256;
  int gMOM = (NN + RPB - 1) / RPB;

  // ---- Block 1: GCN ----
  init_deg  <<<gN, b, 0, stream>>>(deg);
  count_deg <<<gE, b, 0, stream>>>(dstp, deg);
  gemm_wmma_f32<<<gGEMM, b, 0, stream>>>(x, W1, h, NN);
  node_dinv <<<gN, b, 0, stream>>>(deg);
  gcn_init_t<<<gNC, b, 0, stream>>>(h, deg, b1, t);
  gcn_scatter<<<gEW, b, 0, stream>>>(srcp, dstp, h, deg, t);
  zero_moms <<<1, b, 0, stream>>>(sums, sumsq);
  moments   <<<gMOM, dim3(CC), 0, stream>>>(t, nullptr, nullptr, sums, sumsq, RPB);
  norm_out  <<<gNC, b, 0, stream>>>(t, nullptr, nullptr, x, sums, sumsq, gw, gb, gms, xmid);

  // ---- Block 2: GAT ----
  gemm_wmma_f32<<<gGEMM, b, 0, stream>>>(xmid, Wg, h, NN);
  attn_node <<<NN, dim3(CC), 0, stream>>>(h, atts, attd, a_s, a_d, eself, mbuf);
  gat_max   <<<gE, b, 0, stream>>>(srcp, dstp, a_s, a_d, ebuf, mbuf);
  gat_init  <<<gNC, b, 0, stream>>>(h, eself, mbuf, t, den);
  gat_scatter<<<gEW, b, 0, stream>>>(srcp, dstp, h, ebuf, mbuf, t, den);
  zero_moms <<<1, b, 0, stream>>>(sums, sumsq);
  moments   <<<gMOM, dim3(CC), 0, stream>>>(t, den, bg, sums, sumsq, RPB);
  norm_out  <<<gNC, b, 0, stream>>>(t, den, bg, xmid, sums, sumsq, gw, gb, gms, out);
}